// STA_Block_87600152969521
// MI455X (gfx1250) — compile-verified
//
#include <hip/hip_runtime.h>
#include <math.h>

// ---- problem constants (from reference) ----
constexpr int N_  = 16;
constexpr int C_  = 64;
constexpr int T_  = 20;
constexpr int V_  = 150;
constexpr int H_  = 8;
constexpr int D_  = 32;
constexpr int O_  = 128;
constexpr int TV_ = T_ * V_;          // 3000
constexpr int QKV_ = 2 * H_ * D_;     // 512

typedef float v2f __attribute__((ext_vector_type(2)));
typedef float v8f __attribute__((ext_vector_type(8)));

__device__ __forceinline__ int imin(int a, int b) { return a < b ? a : b; }
__device__ __forceinline__ int imax(int a, int b) { return a > b ? a : b; }

// ---------------------------------------------------------------------------
// Kernel 1: per-(c,v) additive table = sinusoidal positional emb + degree emb
// ---------------------------------------------------------------------------
__global__ void build_addcv(const float* __restrict__ degree_emb,
                            const int* __restrict__ all_degree,
                            float* __restrict__ addcv)
{
    int i = blockIdx.x * 256 + threadIdx.x;
    if (i >= C_ * V_) return;
    int c = i / V_;
    int v = i % V_;
    float di = expf(-(float)(c & ~1) * (logf(10000.0f) / (float)C_));
    float pe = (c & 1) ? cosf((float)v * di) : sinf((float)v * di);
    float deg = degree_emb[all_degree[v % 25] * C_ + c];
    addcv[i] = pe + deg;
}

// ---------------------------------------------------------------------------
// Kernel 2: xs = x + addcv[c,v]   (broadcast over n,t)
// ---------------------------------------------------------------------------
__global__ void add_embed(const float* __restrict__ x,
                          const float* __restrict__ addcv,
                          float* __restrict__ xs, long total)
{
    for (long i = blockIdx.x * 256L + threadIdx.x; i < total;
         i += (long)gridDim.x * 256L) {
        int c = (int)((i / TV_) % C_);
        int v = (int)(i % V_);
        xs[i] = x[i] + addcv[c * V_ + v];
    }
}

// ---------------------------------------------------------------------------
// Generic batched GEMM on the fp32 matrix pipe:
//   Y[b][m][n] = sum_k A[b/adiv][m][k] * B[b][k][n] (+ bias[m])
// One wave owns a 16 x (16*NT) strip: A fragment loaded once per K-step and
// reused across NT consecutive WMMAs. All loads are unconditional with
// clamped addresses (garbage lands in rows/cols that are never stored);
// only the K-tail (K%4 != 0) uses selects to inject zeros.
// Requirement: M % 16 == 0 (true for all call sites: 512/128/1280).
// ---------------------------------------------------------------------------
template <int NT>
__global__ __launch_bounds__(256) void gemm_wmma_f32(
    const float* __restrict__ A, int lda, long strideA, int adiv,
    const float* __restrict__ B, int ldb, long strideB,
    const float* __restrict__ bias,
    float* __restrict__ Y, int ldy, long strideY,
    int M, int K, int Ncols)
{
    const int lane = threadIdx.x & 31;
    const int wave = threadIdx.x >> 5;
    const int m0 = blockIdx.y * 128 + wave * 16;
    const int n0 = blockIdx.x * (16 * NT);
    if (m0 >= M) return;                       // wave-uniform (EXEC stays full)
    const int b = blockIdx.z;
    const float* Ab = A + (long)(b / adiv) * strideA;
    const float* Bb = B + (long)b * strideB;
    float* Yb = Y + (long)b * strideY;

    const bool hi  = lane >= 16;
    const int  l16 = lane & 15;
    const int  am  = imin(m0 + l16, M - 1);    // clamp: dead rows never stored
    int bnc[NT];
#pragma unroll
    for (int j = 0; j < NT; ++j) bnc[j] = imin(n0 + 16 * j + l16, Ncols - 1);

    const float* Arow = Ab + (long)am * lda + (hi ? 2 : 0);

    v8f zero = {};
    v8f acc[NT];
#pragma unroll
    for (int j = 0; j < NT; ++j) acc[j] = zero;

    const int K4 = K & ~3;
    for (int k0 = 0; k0 < K4; k0 += 4) {
        v2f a;
        a.x = Arow[k0];
        a.y = Arow[k0 + 1];
        const long kb = (long)(k0 + (hi ? 2 : 0)) * ldb;
        const float* B0p = Bb + kb;
        const float* B1p = Bb + kb + ldb;
#pragma unroll
        for (int j = 0; j < NT; ++j) {
            v2f bb;
            bb.x = B0p[bnc[j]];
            bb.y = B1p[bnc[j]];
            acc[j] = __builtin_amdgcn_wmma_f32_16x16x4_f32(
                false, a, false, bb, (short)0, acc[j], false, false);
        }
    }
    if (K4 < K) {   // peeled guarded K-tail (attention-apply: K=150)
        const int ka = K4 + (hi ? 2 : 0);
        const int kc0 = imin(ka, K - 1);
        const int kc1 = imin(ka + 1, K - 1);
        const bool ok0 = ka < K, ok1 = ka + 1 < K;
        float ax = Ab[(long)am * lda + kc0];
        float ay = Ab[(long)am * lda + kc1];
        v2f a;
        a.x = ok0 ? ax : 0.0f;
        a.y = ok1 ? ay : 0.0f;
        const float* B0p = Bb + (long)kc0 * ldb;
        const float* B1p = Bb + (long)kc1 * ldb;
#pragma unroll
        for (int j = 0; j < NT; ++j) {
            float bx = B0p[bnc[j]];
            float by = B1p[bnc[j]];
            v2f bb;
            bb.x = ok0 ? bx : 0.0f;
            bb.y = ok1 ? by : 0.0f;
            acc[j] = __builtin_amdgcn_wmma_f32_16x16x4_f32(
                false, a, false, bb, (short)0, acc[j], false, false);
        }
    }

    const int rbase = m0 + (hi ? 8 : 0);
#pragma unroll
    for (int j = 0; j < NT; ++j) {
        int col = n0 + 16 * j + l16;
        if (col < Ncols) {
#pragma unroll
            for (int r = 0; r < 8; ++r) {
                int row = rbase + r;
                float bv = bias ? bias[row] : 0.0f;
                Yb[(long)row * ldy + col] = acc[j][r] + bv;
            }
        }
    }
}

// ---------------------------------------------------------------------------
// Attention scores: att[n,h,u,v] = tanh((Q^T K)/(d*T))*alpha[h]
//                                + att0[h,u,v] + spatial_emb[spatial_pos[u,v],h]
// Q,K are (640, V) per (n,h); K=640 (no tail). NT=2 column tiles per wave.
// ---------------------------------------------------------------------------
__global__ __launch_bounds__(256) void att_wmma(
    const float* __restrict__ qkv,
    const float* __restrict__ alphas,
    const float* __restrict__ att0s,
    const float* __restrict__ spatial_emb,
    const int* __restrict__ spatial_pos,
    float* __restrict__ att)
{
    const int lane = threadIdx.x & 31;
    const int wave = threadIdx.x >> 5;
    const int u0 = blockIdx.y * 128 + wave * 16;
    if (u0 >= V_) return;
    const int v0 = blockIdx.x * 32;
    const int b = blockIdx.z;                  // n*H + h
    const int n = b / H_, h = b % H_;
    const float* Q  = qkv + ((long)n * QKV_ + h * D_) * TV_;
    const float* Km = qkv + ((long)n * QKV_ + H_ * D_ + h * D_) * TV_;

    const bool hi  = lane >= 16;
    const int  l16 = lane & 15;
    const int  au  = imin(u0 + l16, V_ - 1);
    const int  bc0 = imin(v0 + l16, V_ - 1);
    const int  bc1 = imin(v0 + 16 + l16, V_ - 1);

    const int hoff = hi ? 2 * V_ : 0;
    const float* Qp  = Q + au + hoff;
    const float* K0p = Km + bc0 + hoff;
    const float* K1p = Km + bc1 + hoff;

    v8f zero = {};
    v8f acc0 = zero, acc1 = zero;
    for (int k0 = 0; k0 < D_ * T_; k0 += 4) {
        const long kb = (long)k0 * V_;
        v2f a, b0, b1;
        a.x  = Qp[kb];        a.y  = Qp[kb + V_];
        b0.x = K0p[kb];       b0.y = K0p[kb + V_];
        b1.x = K1p[kb];       b1.y = K1p[kb + V_];
        acc0 = __builtin_amdgcn_wmma_f32_16x16x4_f32(false, a, false, b0,
                                                     (short)0, acc0, false, false);
        acc1 = __builtin_amdgcn_wmma_f32_16x16x4_f32(false, a, false, b1,
                                                     (short)0, acc1, false, false);
    }

    const float alpha = alphas[h];
    const float inv = 1.0f / (float)(D_ * T_);
    const int rbase = u0 + (hi ? 8 : 0);
#pragma unroll
    for (int j = 0; j < 2; ++j) {
        int bv = v0 + 16 * j + l16;
        if (bv < V_) {
#pragma unroll
            for (int r = 0; r < 8; ++r) {
                int u = rbase + r;
                if (u < V_) {
                    int sp = spatial_pos[u * V_ + bv];
                    float accv = j == 0 ? acc0[r] : acc1[r];
                    float val = tanhf(accv * inv) * alpha
                              + att0s[((long)h * V_ + u) * V_ + bv]
                              + spatial_emb[sp * H_ + h];
                    att[((long)b * V_ + u) * V_ + bv] = val;
                }
            }
        }
    }
}

// ---------------------------------------------------------------------------
// 1x5 conv over V as implicit-im2col GEMM: M=O_=128 (8 waves), K=512*5.
// Padding zeros injected via select on unconditionally-loaded clamped addrs.
// ---------------------------------------------------------------------------
__global__ __launch_bounds__(256) void conv1x5_wmma(
    const float* __restrict__ Xs2,   // (N,512,T,V)
    const float* __restrict__ W,     // (128,512,1,5)
    const float* __restrict__ bias,
    float* __restrict__ Y)           // (N,128,T,V)
{
    const int lane = threadIdx.x & 31;
    const int wave = threadIdx.x >> 5;
    const int m0 = wave * 16;                  // 8 waves cover O_=128
    const int v0 = blockIdx.x * 32;
    const int nt = blockIdx.z;
    const int n = nt / T_, t = nt % T_;

    const bool hi  = lane >= 16;
    const int  l16 = lane & 15;
    const int  am  = m0 + l16;                 // always < 128
    const float* Xb = Xs2 + ((long)n * QKV_ * T_ + t) * V_;
    const float* Wm = W + (long)am * (QKV_ * 5) + (hi ? 10 : 0);

    const int bv0 = v0 + l16, bv1 = v0 + 16 + l16;
    v8f zero = {};
    v8f acc0 = zero, acc1 = zero;

    for (int kk = 0; kk < 5; ++kk) {
        const int vs0 = bv0 + kk - 2, vs1 = bv1 + kk - 2;
        const bool ok0 = vs0 >= 0 && vs0 < V_;
        const bool ok1 = vs1 >= 0 && vs1 < V_;
        const float* X0 = Xb + imax(imin(vs0, V_ - 1), 0);
        const float* X1 = Xb + imax(imin(vs1, V_ - 1), 0);
        const float* Wk = Wm + kk;
        for (int c0 = 0; c0 < QKV_; c0 += 4) {
            v2f a;
            a.x = Wk[(long)c0 * 5];
            a.y = Wk[(long)c0 * 5 + 5];
            const long kb = (long)(c0 + (hi ? 2 : 0)) * TV_;
            float t00 = X0[kb], t01 = X0[kb + TV_];
            float t10 = X1[kb], t11 = X1[kb + TV_];
            v2f b0, b1;
            b0.x = ok0 ? t00 : 0.0f;  b0.y = ok0 ? t01 : 0.0f;
            b1.x = ok1 ? t10 : 0.0f;  b1.y = ok1 ? t11 : 0.0f;
            acc0 = __builtin_amdgcn_wmma_f32_16x16x4_f32(false, a, false, b0,
                                                         (short)0, acc0, false, false);
            acc1 = __builtin_amdgcn_wmma_f32_16x16x4_f32(false, a, false, b1,
                                                         (short)0, acc1, false, false);
        }
    }

    const int rbase = m0 + (hi ? 8 : 0);
#pragma unroll
    for (int j = 0; j < 2; ++j) {
        int bv = j == 0 ? bv0 : bv1;
        if (bv < V_) {
#pragma unroll
            for (int r = 0; r < 8; ++r) {
                int o = rbase + r;
                float accv = j == 0 ? acc0[r] : acc1[r];
                Y[(((long)n * O_ + o) * T_ + t) * V_ + bv] = accv + bias[o];
            }
        }
    }
}

// ---------------------------------------------------------------------------
// 3x1 conv over T as implicit GEMM: M=O_=128, K=O_*3.
// ---------------------------------------------------------------------------
__global__ __launch_bounds__(256) void conv3x1_wmma(
    const float* __restrict__ X,     // (N,128,T,V)
    const float* __restrict__ W,     // (128,128,3,1)
    const float* __restrict__ bias,
    float* __restrict__ Y)           // (N,128,T,V)
{
    const int lane = threadIdx.x & 31;
    const int wave = threadIdx.x >> 5;
    const int m0 = wave * 16;
    const int v0 = blockIdx.x * 32;
    const int nt = blockIdx.z;
    const int n = nt / T_, t = nt % T_;

    const bool hi  = lane >= 16;
    const int  l16 = lane & 15;
    const int  am  = m0 + l16;
    const int  bv0 = v0 + l16, bv1 = v0 + 16 + l16;
    const int  bc0 = imin(bv0, V_ - 1), bc1 = imin(bv1, V_ - 1);
    const float* Wm = W + (long)am * (O_ * 3) + (hi ? 6 : 0);

    v8f zero = {};
    v8f acc0 = zero, acc1 = zero;

    for (int kt = 0; kt < 3; ++kt) {
        const int tt = t + kt - 1;
        const bool tok = tt >= 0 && tt < T_;
        const int ttc = imax(imin(tt, T_ - 1), 0);
        const float* Xb = X + ((long)n * O_ * T_ + ttc) * V_;
        const float* Wk = Wm + kt;
        for (int c0 = 0; c0 < O_; c0 += 4) {
            v2f a;
            a.x = Wk[(long)c0 * 3];
            a.y = Wk[(long)c0 * 3 + 3];
            const long kb = (long)(c0 + (hi ? 2 : 0)) * TV_;
            float t00 = Xb[kb + bc0], t01 = Xb[kb + TV_ + bc0];
            float t10 = Xb[kb + bc1], t11 = Xb[kb + TV_ + bc1];
            v2f b0, b1;
            b0.x = tok ? t00 : 0.0f;  b0.y = tok ? t01 : 0.0f;
            b1.x = tok ? t10 : 0.0f;  b1.y = tok ? t11 : 0.0f;
            acc0 = __builtin_amdgcn_wmma_f32_16x16x4_f32(false, a, false, b0,
                                                         (short)0, acc0, false, false);
            acc1 = __builtin_amdgcn_wmma_f32_16x16x4_f32(false, a, false, b1,
                                                         (short)0, acc1, false, false);
        }
    }

    const int rbase = m0 + (hi ? 8 : 0);
#pragma unroll
    for (int j = 0; j < 2; ++j) {
        int bv = j == 0 ? bv0 : bv1;
        if (bv < V_) {
#pragma unroll
            for (int r = 0; r < 8; ++r) {
                int o = rbase + r;
                float accv = j == 0 ? acc0[r] : acc1[r];
                Y[(((long)n * O_ + o) * T_ + t) * V_ + bv] = accv + bias[o];
            }
        }
    }
}

// ---------------------------------------------------------------------------
// BatchNorm statistics: one workgroup per channel -> scale/shift
// ---------------------------------------------------------------------------
__global__ __launch_bounds__(256) void bn_stats(
    const float* __restrict__ Y, const float* __restrict__ g,
    const float* __restrict__ be, float* __restrict__ scale,
    float* __restrict__ shift, int Cch)
{
    const int o = blockIdx.x;
    double s = 0.0, sq = 0.0;
    for (int n = 0; n < N_; ++n) {
        const float* p = Y + ((long)n * Cch + o) * TV_;
        for (int i = threadIdx.x; i < TV_; i += 256) {
            float v = p[i];
            s += v;
            sq += (double)v * v;
        }
    }
    __shared__ double ss[256], sg[256];
    ss[threadIdx.x] = s; sg[threadIdx.x] = sq;
    __syncthreads();
    for (int st = 128; st > 0; st >>= 1) {
        if (threadIdx.x < st) {
            ss[threadIdx.x] += ss[threadIdx.x + st];
            sg[threadIdx.x] += sg[threadIdx.x + st];
        }
        __syncthreads();
    }
    if (threadIdx.x == 0) {
        double cnt = (double)N_ * TV_;
        double m = ss[0] / cnt;
        double var = sg[0] / cnt - m * m;
        float sc = g[o] * rsqrtf((float)var + 1e-5f);
        scale[o] = sc;
        shift[o] = be[o] - (float)m * sc;
    }
}

// out = leaky_relu(bn(y) + bn(r)); optionally stash bn(r)
__global__ void ew_bn2_lrelu(const float* __restrict__ y,
                             const float* __restrict__ sy, const float* __restrict__ shy,
                             const float* __restrict__ r2,
                             const float* __restrict__ sr, const float* __restrict__ shr,
                             float* __restrict__ rbn, float* __restrict__ out, long total)
{
    for (long i = blockIdx.x * 256L + threadIdx.x; i < total;
         i += (long)gridDim.x * 256L) {
        int o = (int)((i / TV_) % O_);
        float rb = r2[i] * sr[o] + shr[o];
        if (rbn) rbn[i] = rb;
        float v = y[i] * sy[o] + shy[o] + rb;
        out[i] = v >= 0.0f ? v : 0.1f * v;
    }
}

// out = leaky_relu(bn(y) + pre)
__global__ void ew_bn1_add_lrelu(const float* __restrict__ y,
                                 const float* __restrict__ sy, const float* __restrict__ shy,
                                 const float* __restrict__ pre,
                                 float* __restrict__ out, long total)
{
    for (long i = blockIdx.x * 256L + threadIdx.x; i < total;
         i += (long)gridDim.x * 256L) {
        int o = (int)((i / TV_) % O_);
        float v = y[i] * sy[o] + shy[o] + pre[i];
        out[i] = v >= 0.0f ? v : 0.1f * v;
    }
}

// ---------------------------------------------------------------------------
extern "C" void kernel_launch(void* const* d_in, const int* in_sizes, int n_in,
                              void* d_out, int out_size, void* d_ws, size_t ws_size,
                              hipStream_t stream) {
    const float* x           = (const float*)d_in[0];
    const float* W_qkv       = (const float*)d_in[1];
    const float* b_qkv       = (const float*)d_in[2];
    const float* alphas      = (const float*)d_in[3];
    const float* att0s       = (const float*)d_in[4];
    const float* degree_emb  = (const float*)d_in[5];
    const float* spatial_emb = (const float*)d_in[6];
    const float* W_out       = (const float*)d_in[7];
    const float* b_out       = (const float*)d_in[8];
    const float* g_out       = (const float*)d_in[9];
    const float* be_out      = (const float*)d_in[10];
    const float* W_ff        = (const float*)d_in[11];
    const float* b_ff        = (const float*)d_in[12];
    const float* g_ff        = (const float*)d_in[13];
    const float* be_ff       = (const float*)d_in[14];
    const float* W_t         = (const float*)d_in[15];
    const float* b_t         = (const float*)d_in[16];
    const float* g_t         = (const float*)d_in[17];
    const float* be_t        = (const float*)d_in[18];
    const float* W_rs        = (const float*)d_in[19];
    const float* b_rs        = (const float*)d_in[20];
    const float* g_rs        = (const float*)d_in[21];
    const float* be_rs       = (const float*)d_in[22];
    const float* W_rt        = (const float*)d_in[23];
    const float* b_rt        = (const float*)d_in[24];
    const float* g_rt        = (const float*)d_in[25];
    const float* be_rt       = (const float*)d_in[26];
    const int*   all_degree  = (const int*)d_in[27];
    const int*   spatial_pos = (const int*)d_in[28];
    float* out = (float*)d_out;

    // ---- workspace carve-up (floats), with buffer recycling ----
    float* ws = (float*)d_ws;
    size_t off = 0;
    auto alloc = [&](size_t nf) { float* p = ws + off; off += nf; return p; };
    float* addcv = alloc((size_t)C_ * V_);                 // 9,600
    float* xs    = alloc((size_t)N_ * C_ * TV_);           // 3.07M
    float* qkv   = alloc((size_t)N_ * QKV_ * TV_);         // 24.58M (later xs2)
    float* att   = alloc((size_t)N_ * H_ * V_ * V_);       // 2.88M
    float* B0    = alloc((size_t)N_ * O_ * TV_);           // y0 -> y1 -> y2
    float* B1    = alloc((size_t)N_ * O_ * TV_);           // r  -> xs_b
    float* B2    = alloc((size_t)N_ * O_ * TV_);           // xs_a -> y3
    float* B3    = alloc((size_t)N_ * O_ * TV_);           // rbn
    float* stats = alloc(2048);
    (void)ws_size; (void)in_sizes; (void)n_in; (void)out_size;

    float* s_out = stats;        float* sh_out = stats + 128;
    float* s_rs  = stats + 256;  float* sh_rs  = stats + 384;
    float* s_ff  = stats + 512;  float* sh_ff  = stats + 640;
    float* s_t   = stats + 768;  float* sh_t   = stats + 896;
    float* s_rt  = stats + 1024; float* sh_rt  = stats + 1152;

    const long totO = (long)N_ * O_ * TV_;   // 6.144M
    const long totC = (long)N_ * C_ * TV_;   // 3.072M
    const int  gx64 = (TV_ + 63) / 64;       // 47 column-blocks (NT=4)
    const int  gx32v = (V_ + 31) / 32;       // 5 column-blocks (NT=2)

    // S1/S2: embeddings
    build_addcv<<<(C_ * V_ + 255) / 256, 256, 0, stream>>>(degree_emb, all_degree, addcv);
    add_embed<<<4096, 256, 0, stream>>>(x, addcv, xs, totC);

    // S3: qkv projection  (M=512, K=64, N=3000, batch=N)
    gemm_wmma_f32<4><<<dim3(gx64, 4, N_), 256, 0, stream>>>(
        W_qkv, C_, 0L, 1,
        xs, TV_, (long)C_ * TV_,
        b_qkv,
        qkv, TV_, (long)QKV_ * TV_,
        QKV_, C_, TV_);

    // S4: attention scores with tanh/alpha/bias epilogue
    att_wmma<<<dim3(gx32v, 2, N_ * H_), 256, 0, stream>>>(
        qkv, alphas, att0s, spatial_emb, spatial_pos, att);

    // S5: attention apply: xs2[(n,h),(c,t),v] = x[n,(c,t),u] @ att[(n,h),u,v]
    float* xs2 = qkv;  // qkv buffer no longer needed
    gemm_wmma_f32<2><<<dim3(gx32v, (C_ * T_) / 128, N_ * H_), 256, 0, stream>>>(
        x, V_, (long)C_ * TV_, H_,
        att, V_, (long)V_ * V_,
        nullptr,
        xs2, V_, (long)C_ * TV_,
        C_ * T_, V_, V_);

    // S6: W_out 1x5 conv (heaviest GEMM, K=2560)
    float* y0 = B0;
    conv1x5_wmma<<<dim3(gx32v, 1, N_ * T_), 256, 0, stream>>>(xs2, W_out, b_out, y0);

    // S7: residual rs = 1x1 conv of x
    float* r = B1;
    gemm_wmma_f32<4><<<dim3(gx64, 1, N_), 256, 0, stream>>>(
        W_rs, C_, 0L, 1,
        x, TV_, (long)C_ * TV_,
        b_rs,
        r, TV_, (long)O_ * TV_,
        O_, C_, TV_);

    // S8: BN stats, then fused bn+bn+add+leaky
    bn_stats<<<O_, 256, 0, stream>>>(y0, g_out, be_out, s_out, sh_out, O_);
    bn_stats<<<O_, 256, 0, stream>>>(r, g_rs, be_rs, s_rs, sh_rs, O_);
    float* rbn  = B3;
    float* xs_a = B2;
    ew_bn2_lrelu<<<4096, 256, 0, stream>>>(y0, s_out, sh_out, r, s_rs, sh_rs,
                                           rbn, xs_a, totO);

    // S10: ff 1x1 conv
    float* y1 = B0;
    gemm_wmma_f32<4><<<dim3(gx64, 1, N_), 256, 0, stream>>>(
        W_ff, O_, 0L, 1,
        xs_a, TV_, (long)O_ * TV_,
        b_ff,
        y1, TV_, (long)O_ * TV_,
        O_, O_, TV_);
    bn_stats<<<O_, 256, 0, stream>>>(y1, g_ff, be_ff, s_ff, sh_ff, O_);
    float* xs_b = B1;
    ew_bn1_add_lrelu<<<4096, 256, 0, stream>>>(y1, s_ff, sh_ff, rbn, xs_b, totO);

    // S12: temporal 3x1 conv + rt 1x1 conv
    float* y2 = B0;
    conv3x1_wmma<<<dim3(gx32v, 1, N_ * T_), 256, 0, stream>>>(xs_b, W_t, b_t, y2);
    float* y3 = B2;
    gemm_wmma_f32<4><<<dim3(gx64, 1, N_), 256, 0, stream>>>(
        W_rt, O_, 0L, 1,
        xs_b, TV_, (long)O_ * TV_,
        b_rt,
        y3, TV_, (long)O_ * TV_,
        O_, O_, TV_);

    // S13: final bn+bn+add+leaky -> d_out
    bn_stats<<<O_, 256, 0, stream>>>(y2, g_t, be_t, s_t, sh_t, O_);
    bn_stats<<<O_, 256, 0, stream>>>(y3, g_rt, be_rt, s_rt, sh_rt, O_);
    ew_bn2_lrelu<<<4096, 256, 0, stream>>>(y2, s_t, sh_t, y3, s_rt, sh_rt,
                                           nullptr, out, totO);
}